// Model_28862180229813
// MI455X (gfx1250) — compile-verified
//
#include <hip/hip_runtime.h>
#include <hip/hip_bf16.h>

typedef __attribute__((ext_vector_type(2))) float v2f;
typedef __attribute__((ext_vector_type(8))) float v8f;

#define Bn 8
#define Tn 12
#define Nn 512
#define Hn 32
#define COn 64

// ---------- helpers: half-wave (16-lane) reductions, matching WMMA C-layout rows ----------
__device__ __forceinline__ float hmax16(float v) {
    #pragma unroll
    for (int m = 8; m; m >>= 1) v = fmaxf(v, __shfl_xor(v, m, 16));
    return v;
}
__device__ __forceinline__ float hsum16(float v) {
    #pragma unroll
    for (int m = 8; m; m >>= 1) v += __shfl_xor(v, m, 16);
    return v;
}

// ---------- fc_in: (B,T,N,2) -> h (B,32,N,12) ----------
__global__ void fcin_kernel(const float* __restrict__ x, const float* __restrict__ w,
                            const float* __restrict__ bias, float* __restrict__ h) {
    int idx = blockIdx.x * blockDim.x + threadIdx.x;  // (b,n,t)
    if (idx >= Bn * Nn * Tn) return;
    int t = idx % Tn;
    int n = (idx / Tn) % Nn;
    int b = idx / (Tn * Nn);
    float x0 = x[(((size_t)b * Tn + t) * Nn + n) * 2 + 0];
    float x1 = x[(((size_t)b * Tn + t) * Nn + n) * 2 + 1];
    #pragma unroll
    for (int c = 0; c < Hn; ++c)
        h[(((size_t)b * Hn + c) * Nn + n) * Tn + t] = w[c * 2] * x0 + w[c * 2 + 1] * x1 + bias[c];
}

// ---------- K/Q/V temporal conv: h (B,32,N,12) -> (B,12,N,D) each ----------
template <int TF>
__global__ void kqv_kernel(const float* __restrict__ h,
                           const float* __restrict__ kw, const float* __restrict__ kb,
                           const float* __restrict__ qw, const float* __restrict__ qb,
                           const float* __restrict__ vw, const float* __restrict__ vb,
                           float* __restrict__ K, float* __restrict__ Q, float* __restrict__ V) {
    constexpr int D = Tn - TF + 1;
    int idx = blockIdx.x * blockDim.x + threadIdx.x;  // ((b*12+tq)*512+n)*D + c
    if (idx >= Bn * Tn * Nn * D) return;
    int c  = idx % D;
    int n  = (idx / D) % Nn;
    int tq = (idx / (D * Nn)) % Tn;
    int b  = idx / (D * Nn * Tn);
    float sk = kb[tq], sq = qb[tq], sv = vb[tq];
    for (int hh = 0; hh < Hn; ++hh) {
        const float* hp  = h + (((size_t)b * Hn + hh) * Nn + n) * Tn + c;
        const float* kwp = kw + (tq * Hn + hh) * TF;
        const float* qwp = qw + (tq * Hn + hh) * TF;
        const float* vwp = vw + (tq * Hn + hh) * TF;
        #pragma unroll
        for (int dt = 0; dt < TF; ++dt) {
            float xv = hp[dt];
            sk += kwp[dt] * xv; sq += qwp[dt] * xv; sv += vwp[dt] * xv;
        }
    }
    K[idx] = sk; Q[idx] = sq; V[idx] = sv;
}

// ---------- fused flash attention with V_WMMA_F32_16X16X4_F32 ----------
// grid (8, 96): 4 waves/block, each wave owns one 16-row tile of one (b,t) problem.
template <int D>
__global__ __launch_bounds__(128) void attn_kernel(
    const float* __restrict__ Kt, const float* __restrict__ Qt, const float* __restrict__ Vt,
    const float* __restrict__ adj, const float* __restrict__ adjh, float* __restrict__ att) {
    constexpr int NK = (D + 3) / 4;
    const int lane = threadIdx.x & 31;
    const int wave = threadIdx.x >> 5;
    const int l16  = lane & 15;
    const int half = lane >> 4;       // lanes 0-15: rows r / K 0,1 ; lanes 16-31: rows r+8 / K 2,3
    const int i0   = (blockIdx.x * 4 + wave) * 16;
    const int b    = blockIdx.y / Tn;
    const int t    = blockIdx.y % Tn;

    const float* Kbt = Kt + (size_t)(b * Tn + t) * Nn * D;
    const float* Qbt = Qt + (size_t)(b * Tn + t) * Nn * D;
    const float* Vbt = Vt + (size_t)(b * Tn + t) * Nn * D;
    const float* A1  = adj  + (size_t)b * Nn * Nn;
    const float* A2  = adjh + (size_t)b * Nn * Nn;

    // A operand (16x4 chunks of K rows), loaded once
    v2f aK[NK];
    #pragma unroll
    for (int kc = 0; kc < NK; ++kc)
        #pragma unroll
        for (int u = 0; u < 2; ++u) {
            int kk = kc * 4 + half * 2 + u;
            aK[kc][u] = (kk < D) ? Kbt[(i0 + l16) * D + kk] : 0.0f;
        }

    float m[8], l[8];
    v8f acc = {};
    #pragma unroll
    for (int r = 0; r < 8; ++r) { m[r] = -1e30f; l[r] = 0.0f; }

    __shared__ float plds[4][16][17];  // stride 17 avoids bank conflicts on transposed read

    for (int jt = 0; jt < Nn / 16; ++jt) {
        int j0 = jt * 16;
        // C initialized with the additive adjacency bias (C-layout)
        v8f s;
        #pragma unroll
        for (int r = 0; r < 8; ++r) {
            size_t o = (size_t)(i0 + r + half * 8) * Nn + (j0 + l16);
            s[r] = A1[o] + A2[o];
        }
        // S = K * Q^T + bias
        #pragma unroll
        for (int kc = 0; kc < NK; ++kc) {
            v2f bQ;
            #pragma unroll
            for (int u = 0; u < 2; ++u) {
                int kk = kc * 4 + half * 2 + u;
                bQ[u] = (kk < D) ? Qbt[(j0 + l16) * D + kk] : 0.0f;
            }
            s = __builtin_amdgcn_wmma_f32_16x16x4_f32(false, aK[kc], false, bQ,
                                                      (short)0, s, false, false);
        }
        // online softmax (per-row stats live in the same half-wave striping as C-layout)
        #pragma unroll
        for (int r = 0; r < 8; ++r) {
            float rm = hmax16(s[r]);
            float nm = fmaxf(m[r], rm);
            float sc = __expf(m[r] - nm);
            s[r] = __expf(s[r] - nm);
            l[r] = l[r] * sc + hsum16(s[r]);
            acc[r] *= sc;
            m[r] = nm;
        }
        // transpose P (C-layout -> A-layout) through LDS
        __syncthreads();
        #pragma unroll
        for (int r = 0; r < 8; ++r) plds[wave][r + half * 8][l16] = s[r];
        __syncthreads();
        // O += P * V   (K dim = 16 columns of this tile -> 4 wmma steps)
        #pragma unroll
        for (int kc = 0; kc < 4; ++kc) {
            v2f aP, bV;
            #pragma unroll
            for (int u = 0; u < 2; ++u) {
                int jl = kc * 4 + half * 2 + u;
                aP[u] = plds[wave][l16][jl];
                bV[u] = (l16 < D) ? Vbt[(size_t)(j0 + jl) * D + l16] : 0.0f;
            }
            acc = __builtin_amdgcn_wmma_f32_16x16x4_f32(false, aP, false, bV,
                                                        (short)0, acc, false, false);
        }
    }
    // normalize rows and store att[b][d][i][t]
    if (l16 < D) {
        #pragma unroll
        for (int r = 0; r < 8; ++r) {
            int row = i0 + r + half * 8;
            att[(((size_t)b * D + l16) * Nn + row) * Tn + t] = acc[r] / l[r];
        }
    }
}

// ---------- ga output conv + residual -> concat slice (pre-BN) ----------
template <int D>
__global__ void gaout_kernel(const float* __restrict__ att, const float* __restrict__ h,
                             const float* __restrict__ ow, const float* __restrict__ ob,
                             const float* __restrict__ rw, const float* __restrict__ rb,
                             float* __restrict__ cc, int ch0) {
    int idx = blockIdx.x * blockDim.x + threadIdx.x;  // (b,n,t)
    if (idx >= Bn * Nn * Tn) return;
    int t = idx % Tn;
    int n = (idx / Tn) % Nn;
    int b = idx / (Tn * Nn);
    float av[D];
    #pragma unroll
    for (int c = 0; c < D; ++c) av[c] = att[(((size_t)b * D + c) * Nn + n) * Tn + t];
    float hv[Hn];
    #pragma unroll
    for (int hh = 0; hh < Hn; ++hh) hv[hh] = h[(((size_t)b * Hn + hh) * Nn + n) * Tn + t];
    for (int co = 0; co < COn; ++co) {
        float s = ob[co] + rb[co];
        #pragma unroll
        for (int c = 0; c < D; ++c) s += ow[co * D + c] * av[c];
        #pragma unroll
        for (int hh = 0; hh < Hn; ++hh) s += rw[co * Hn + hh] * hv[hh];
        cc[(((size_t)b * 256 + ch0 + co) * Nn + n) * Tn + t] = s;
    }
}

// ---------- batchnorm stats: one block per channel (deterministic, no atomics) ----------
__global__ void bnstats_kernel(const float* __restrict__ y, float* __restrict__ stats,
                               int cstride, int ch0) {
    int co = blockIdx.x;
    __shared__ float ssum[256], ssq[256];
    float s = 0.f, q = 0.f;
    for (int e = threadIdx.x; e < Bn * Nn * Tn; e += blockDim.x) {
        int t = e % Tn; int n = (e / Tn) % Nn; int b = e / (Tn * Nn);
        float v = y[(((size_t)b * cstride + ch0 + co) * Nn + n) * Tn + t];
        s += v; q += v * v;
    }
    ssum[threadIdx.x] = s; ssq[threadIdx.x] = q;
    __syncthreads();
    for (int off = 128; off; off >>= 1) {
        if ((int)threadIdx.x < off) {
            ssum[threadIdx.x] += ssum[threadIdx.x + off];
            ssq[threadIdx.x]  += ssq[threadIdx.x + off];
        }
        __syncthreads();
    }
    if (threadIdx.x == 0) {
        const float M = (float)(Bn * Nn * Tn);
        float mean = ssum[0] / M;
        float var  = ssq[0] / M - mean * mean;
        stats[co * 2]     = mean;
        stats[co * 2 + 1] = rsqrtf(var + 1e-5f);
    }
}

__global__ void bnapply_kernel(float* __restrict__ y, const float* __restrict__ stats,
                               const float* __restrict__ g, const float* __restrict__ beta,
                               int cstride, int ch0) {
    int idx = blockIdx.x * blockDim.x + threadIdx.x;  // over B*64*N*T
    if (idx >= Bn * COn * Nn * Tn) return;
    int t  = idx % Tn;
    int n  = (idx / Tn) % Nn;
    int co = (idx / (Tn * Nn)) % COn;
    int b  = idx / (Tn * Nn * COn);
    size_t o = (((size_t)b * cstride + ch0 + co) * Nn + n) * Tn + t;
    y[o] = (y[o] - stats[co * 2]) * stats[co * 2 + 1] * g[co] + beta[co];
}

// ---------- block output conv (256->64) + residual ----------
__global__ void blockout_kernel(const float* __restrict__ cc, const float* __restrict__ h,
                                const float* __restrict__ ow, const float* __restrict__ ob,
                                const float* __restrict__ rw, const float* __restrict__ rb,
                                float* __restrict__ y2) {
    int idx = blockIdx.x * blockDim.x + threadIdx.x;  // (b,n,t) * 4 groups of 16 co
    if (idx >= 4 * Bn * Nn * Tn) return;
    int grp = idx & 3;
    int e   = idx >> 2;
    int t = e % Tn; int n = (e / Tn) % Nn; int b = e / (Tn * Nn);
    int co0 = grp * 16;
    float acc[16];
    #pragma unroll
    for (int i = 0; i < 16; ++i) acc[i] = ob[co0 + i] + rb[co0 + i];
    for (int k = 0; k < 256; ++k) {
        float v = cc[(((size_t)b * 256 + k) * Nn + n) * Tn + t];
        #pragma unroll
        for (int i = 0; i < 16; ++i) acc[i] += ow[(co0 + i) * 256 + k] * v;
    }
    for (int hh = 0; hh < Hn; ++hh) {
        float v = h[(((size_t)b * Hn + hh) * Nn + n) * Tn + t];
        #pragma unroll
        for (int i = 0; i < 16; ++i) acc[i] += rw[(co0 + i) * Hn + hh] * v;
    }
    #pragma unroll
    for (int i = 0; i < 16; ++i)
        y2[(((size_t)b * COn + co0 + i) * Nn + n) * Tn + t] = acc[i];
}

// ---------- fc_out + transpose to (B,T,N,1) ----------
__global__ void fcout_kernel(const float* __restrict__ y2, const float* __restrict__ w,
                             const float* __restrict__ bias, float* __restrict__ out) {
    int idx = blockIdx.x * blockDim.x + threadIdx.x;  // (b,t,n)
    if (idx >= Bn * Tn * Nn) return;
    int n = idx % Nn; int t = (idx / Nn) % Tn; int b = idx / (Nn * Tn);
    float s = bias[0];
    for (int co = 0; co < COn; ++co)
        s += w[co] * y2[(((size_t)b * COn + co) * Nn + n) * Tn + t];
    out[((size_t)b * Tn + t) * Nn + n] = s;
}

// ---------- host-side per-filter pipeline ----------
template <int TF, int D>
static void run_ga(void* const* d_in, int base, int slot, const float* h,
                   const float* adj, const float* adjh, float* K, float* Q, float* V,
                   float* att, float* cc, float* st, hipStream_t stream) {
    const float* kw = (const float*)d_in[base + 0]; const float* kb = (const float*)d_in[base + 1];
    const float* qw = (const float*)d_in[base + 2]; const float* qb = (const float*)d_in[base + 3];
    const float* vw = (const float*)d_in[base + 4]; const float* vb = (const float*)d_in[base + 5];
    const float* rw = (const float*)d_in[base + 6]; const float* rb = (const float*)d_in[base + 7];
    const float* ow = (const float*)d_in[base + 8]; const float* ob = (const float*)d_in[base + 9];
    const float* g  = (const float*)d_in[base + 10]; const float* be = (const float*)d_in[base + 11];
    int totkqv = Bn * Tn * Nn * D;
    kqv_kernel<TF><<<(totkqv + 255) / 256, 256, 0, stream>>>(h, kw, kb, qw, qb, vw, vb, K, Q, V);
    attn_kernel<D><<<dim3(8, Bn * Tn), 128, 0, stream>>>(K, Q, V, adj, adjh, att);
    gaout_kernel<D><<<(Bn * Nn * Tn + 255) / 256, 256, 0, stream>>>(att, h, ow, ob, rw, rb, cc, slot * 64);
    bnstats_kernel<<<64, 256, 0, stream>>>(cc, st, 256, slot * 64);
    bnapply_kernel<<<(Bn * COn * Nn * Tn + 255) / 256, 256, 0, stream>>>(cc, st, g, be, 256, slot * 64);
}

extern "C" void kernel_launch(void* const* d_in, const int* in_sizes, int n_in,
                              void* d_out, int out_size, void* d_ws, size_t ws_size,
                              hipStream_t stream) {
    // insertion-order flattening of setup_inputs():
    // 0:x 1:idx(unused) 2:adj 3:adj_hat 4:fc_in.w 5:fc_in.b
    // 6..17: ga2{kw,kb,qw,qb,vw,vb,rw,rb,ow,ob,g,beta} 18..29: ga3  30..41: ga6  42..53: ga7
    // 54:block.rw 55:block.rb 56:block.ow 57:block.ob 58:block.g 59:block.beta 60:fc_out.w 61:fc_out.b
    const float* x    = (const float*)d_in[0];
    const float* adj  = (const float*)d_in[2];
    const float* adjh = (const float*)d_in[3];
    const float* fiw  = (const float*)d_in[4];
    const float* fib  = (const float*)d_in[5];
    const float* brw  = (const float*)d_in[54];
    const float* brb  = (const float*)d_in[55];
    const float* bow  = (const float*)d_in[56];
    const float* bob  = (const float*)d_in[57];
    const float* bg   = (const float*)d_in[58];
    const float* bbe  = (const float*)d_in[59];
    const float* fow  = (const float*)d_in[60];
    const float* fob  = (const float*)d_in[61];
    float* out = (float*)d_out;

    // workspace carve-up (floats)
    float* ws  = (float*)d_ws;
    float* h   = ws;                        // 8*32*512*12      = 1,572,864
    float* K   = h + 1572864;               // max 8*12*512*11  =   540,672 each
    float* Q   = K + 540672;
    float* V   = Q + 540672;
    float* att = V + 540672;                // max 540,672 (reused per tf)
    float* cc  = att + 540672;              // 8*256*512*12     = 12,582,912
    float* y2  = cc + 12582912;             // 8*64*512*12      =  3,145,728
    float* st  = y2 + 3145728;              // 128 floats of BN stats

    fcin_kernel<<<(Bn * Nn * Tn + 255) / 256, 256, 0, stream>>>(x, fiw, fib, h);

    run_ga<2, 11>(d_in, 6,  0, h, adj, adjh, K, Q, V, att, cc, st, stream);
    run_ga<3, 10>(d_in, 18, 1, h, adj, adjh, K, Q, V, att, cc, st, stream);
    run_ga<6, 7 >(d_in, 30, 2, h, adj, adjh, K, Q, V, att, cc, st, stream);
    run_ga<7, 6 >(d_in, 42, 3, h, adj, adjh, K, Q, V, att, cc, st, stream);

    blockout_kernel<<<(4 * Bn * Nn * Tn + 255) / 256, 256, 0, stream>>>(cc, h, bow, bob, brw, brb, y2);
    bnstats_kernel<<<64, 256, 0, stream>>>(y2, st, COn, 0);
    bnapply_kernel<<<(Bn * COn * Nn * Tn + 255) / 256, 256, 0, stream>>>(y2, st, bg, bbe, COn, 0);
    fcout_kernel<<<(Bn * Tn * Nn + 255) / 256, 256, 0, stream>>>(y2, fow, fob, out);
}